// GCNStack_87686052315402
// MI455X (gfx1250) — compile-verified
//
#include <hip/hip_runtime.h>

typedef __attribute__((ext_vector_type(2))) float v2f;
typedef __attribute__((ext_vector_type(8))) float v8f;

// ---------------------------------------------------------------------------
// K1: degree count (self-loop added later as +1)
__global__ void k_deg(const int* __restrict__ dst, float* __restrict__ cnt, int E) {
    int e = blockIdx.x * blockDim.x + threadIdx.x;
    if (e < E) atomicAdd(&cnt[dst[e]], 1.0f);
}

// K2: dinv = rsqrt(deg + 1)   (deg >= 1 always due to self loop)
__global__ void k_dinv(const float* __restrict__ cnt, float* __restrict__ dinv, int n) {
    int i = blockIdx.x * blockDim.x + threadIdx.x;
    if (i < n) dinv[i] = rsqrtf(cnt[i] + 1.0f);
}

// K3: layer-1 scalar aggregation: acc1[dst] += dinv[src]*feat[src]
__global__ void k_agg1(const int* __restrict__ src, const int* __restrict__ dst,
                       const float* __restrict__ feat, const float* __restrict__ dinv,
                       float* __restrict__ acc1, int E) {
    int e = blockIdx.x * blockDim.x + threadIdx.x;
    if (e < E) {
        int s = src[e];
        atomicAdd(&acc1[dst[e]], dinv[s] * feat[s]);
    }
}

// K4: aggf[i] = dinv[i]*(acc1[i] + dinv[i]*feat[i])   (self loop folded in)
__global__ void k_fin1(const float* __restrict__ acc1, const float* __restrict__ feat,
                       const float* __restrict__ dinv, float* __restrict__ aggf, int n) {
    int i = blockIdx.x * blockDim.x + threadIdx.x;
    if (i < n) {
        float d = dinv[i];
        aggf[i] = d * (acc1[i] + d * feat[i]);
    }
}

// K5: layer-2 edge scatter. One wave per edge-slice; lane j = output channel.
// x1[src,j] = relu(aggf[src]*W1[j]+b1[j]) recomputed on the fly (no x1 buffer).
// acc2[dst*32+j] += dinv[src] * x1[src,j]   -> 128B coalesced atomic per edge.
__global__ void k_agg2(const int* __restrict__ src, const int* __restrict__ dst,
                       const float* __restrict__ aggf, const float* __restrict__ dinv,
                       const float* __restrict__ W1, const float* __restrict__ b1,
                       float* __restrict__ acc2, int E, int nWaves) {
    int tid  = blockIdx.x * blockDim.x + threadIdx.x;
    int wave = tid >> 5;
    int j    = threadIdx.x & 31;
    float w  = W1[j];
    float bb = b1[j];
    for (int e = wave; e < E; e += nWaves) {
        int   s  = src[e];
        int   d  = dst[e];
        float a  = aggf[s];
        float dv = dinv[s];
        float v  = dv * fmaxf(fmaf(a, w, bb), 0.0f);   // dv > 0, safe to scale after relu
        atomicAdd(&acc2[(size_t)d * 32 + j], v);
    }
}

// K6: finalize layer-2 aggregation in place:
// acc2[i,j] = dinv[i]*(acc2[i,j] + dinv[i]*relu(aggf[i]*W1[j]+b1[j]))
__global__ void k_fin2(float* __restrict__ acc2, const float* __restrict__ aggf,
                       const float* __restrict__ dinv, const float* __restrict__ W1,
                       const float* __restrict__ b1, int n) {
    int t = blockIdx.x * blockDim.x + threadIdx.x;
    if (t < n * 32) {
        int   i  = t >> 5;
        int   j  = t & 31;
        float d  = dinv[i];
        float x1 = fmaxf(fmaf(aggf[i], W1[j], b1[j]), 0.0f);
        acc2[t]  = d * (acc2[t] + d * x1);
    }
}

// K7: out[100000,64] = relu(acc2[100000,32] @ W2[32,64] + b2) via V_WMMA_F32_16X16X4_F32.
// One wave = one 16-row M tile x all four 16-col N tiles. K=32 -> 8 WMMA K4 steps,
// A fragment reused across the 4 N accumulators (32 WMMAs per wave).
__global__ void k_gemm_wmma(const float* __restrict__ X, const float* __restrict__ W2,
                            const float* __restrict__ b2, float* __restrict__ out, int n) {
    int tid    = blockIdx.x * blockDim.x + threadIdx.x;
    int wave   = tid >> 5;                // m-tile index
    int lane   = threadIdx.x & 31;
    int mtiles = n >> 4;                  // n is a multiple of 16 (100000/16 = 6250)
    if (wave >= mtiles) return;           // wave-uniform exit: EXEC stays all-ones

    int m0   = lane & 15;                 // row within tile (A) / col within tile (B,D)
    int kh   = lane >> 4;                 // half-wave selects K sub-pair
    int row  = (wave << 4) + m0;
    row      = row < n ? row : n - 1;     // defensive clamp (select, not EXEC)

    v8f c0 = {}, c1 = {}, c2 = {}, c3 = {};
    const float* xrow = X + (size_t)row * 32;

#pragma unroll
    for (int k = 0; k < 8; ++k) {
        int kb = 4 * k + 2 * kh;          // K columns {kb, kb+1} for this half-wave
        v2f a;
        a.x = xrow[kb];
        a.y = xrow[kb + 1];

        const float* wp = W2 + (size_t)kb * 64;   // rows kb and kb+1 of W2
        v2f b0v, b1v, b2v, b3v;
        b0v.x = wp[m0];          b0v.y = wp[64 + m0];
        b1v.x = wp[16 + m0];     b1v.y = wp[64 + 16 + m0];
        b2v.x = wp[32 + m0];     b2v.y = wp[64 + 32 + m0];
        b3v.x = wp[48 + m0];     b3v.y = wp[64 + 48 + m0];

        c0 = __builtin_amdgcn_wmma_f32_16x16x4_f32(false, a, false, b0v, (short)0, c0, false, false);
        c1 = __builtin_amdgcn_wmma_f32_16x16x4_f32(false, a, false, b1v, (short)0, c1, false, false);
        c2 = __builtin_amdgcn_wmma_f32_16x16x4_f32(false, a, false, b2v, (short)0, c2, false, false);
        c3 = __builtin_amdgcn_wmma_f32_16x16x4_f32(false, a, false, b3v, (short)0, c3, false, false);
    }

    // D layout: VGPR r, lanes 0-15 -> M=r, lanes 16-31 -> M=8+r; N = lane&15.
    float bb0 = b2[m0], bb1 = b2[16 + m0], bb2 = b2[32 + m0], bb3 = b2[48 + m0];
    int rbase = (wave << 4) + 8 * kh;
#pragma unroll
    for (int r = 0; r < 8; ++r) {
        int orow = rbase + r;
        orow = orow < n ? orow : n - 1;
        float* op = out + (size_t)orow * 64;
        op[m0]      = fmaxf(c0[r] + bb0, 0.0f);
        op[16 + m0] = fmaxf(c1[r] + bb1, 0.0f);
        op[32 + m0] = fmaxf(c2[r] + bb2, 0.0f);
        op[48 + m0] = fmaxf(c3[r] + bb3, 0.0f);
    }
}

// ---------------------------------------------------------------------------
extern "C" void kernel_launch(void* const* d_in, const int* in_sizes, int n_in,
                              void* d_out, int out_size, void* d_ws, size_t ws_size,
                              hipStream_t stream) {
    const float* feat  = (const float*)d_in[0];
    const int*   edges = (const int*)d_in[1];   // [2, E] int32
    const float* W1    = (const float*)d_in[2];
    const float* b1    = (const float*)d_in[3];
    const float* W2    = (const float*)d_in[4];
    const float* b2    = (const float*)d_in[5];
    float*       out   = (float*)d_out;

    int n = in_sizes[0];
    int E = in_sizes[1] / 2;
    const int* src = edges;
    const int* dst = edges + E;

    // Workspace layout (floats): cnt[n] | dinv[n] | acc1[n] | aggf[n] | acc2[32n]
    float* ws   = (float*)d_ws;
    float* cnt  = ws;
    float* dinv = ws + (size_t)n;
    float* acc1 = ws + 2 * (size_t)n;
    float* aggf = ws + 3 * (size_t)n;
    float* acc2 = ws + 4 * (size_t)n;

    hipMemsetAsync(cnt,  0, (size_t)n * sizeof(float), stream);
    hipMemsetAsync(acc1, 0, (size_t)n * sizeof(float), stream);
    hipMemsetAsync(acc2, 0, (size_t)n * 32 * sizeof(float), stream);

    const int B = 256;                      // 8 wave32s per block
    k_deg <<<(E + B - 1) / B, B, 0, stream>>>(dst, cnt, E);
    k_dinv<<<(n + B - 1) / B, B, 0, stream>>>(cnt, dinv, n);
    k_agg1<<<(E + B - 1) / B, B, 0, stream>>>(src, dst, feat, dinv, acc1, E);
    k_fin1<<<(n + B - 1) / B, B, 0, stream>>>(acc1, feat, dinv, aggf, n);

    const int blocks2 = 2048;
    const int nWaves  = blocks2 * (B / 32);
    k_agg2<<<blocks2, B, 0, stream>>>(src, dst, aggf, dinv, W1, b1, acc2, E, nWaves);

    k_fin2<<<((size_t)n * 32 + B - 1) / B, B, 0, stream>>>(acc2, aggf, dinv, W1, b1, n);

    int mtiles  = (n + 15) / 16;            // one wave per 16-row tile
    int blocks3 = (mtiles * 32 + B - 1) / B;
    k_gemm_wmma<<<blocks3, B, 0, stream>>>(acc2, W2, b2, out, n);
}